// DistributedKnowledgeCongruence_56410100465967
// MI455X (gfx1250) — compile-verified
//
#include <hip/hip_runtime.h>
#include <stdint.h>

// DistributedKnowledgeCongruence on MI455X (gfx1250).
//
// Roofline: 131072x1000 f32 in + same out = ~1.05 GB of HBM traffic, ~0.5 GFLOP
// => pure bandwidth problem, floor ~45us at 23.3 TB/s. No contraction exists, so
// WMMA is not applicable; the CDNA5-specific win is the async global->LDS DMA
// path (GLOBAL_LOAD_ASYNC_TO_LDS_*, ASYNCcnt) for double-buffered row staging,
// plus non-temporal cache hints: 1.05 GB streamed once through a 192 MB L2 has
// zero reuse, so both the input loads and output stores are marked NT.
//
// Math: new_k = (899*x + max - 0.9) / (1000*max - 1) is affine in x, so
// min(new_k) = min(new_k(row_min), new_k(row_max)). One pass for
// (max, min, argmax), one pass to emit either the rescale or the fallback.

namespace {
constexpr int   kC     = 1000;   // classes per row
constexpr float kT     = 0.9f;   // congruent max probability
constexpr int   kRowF4 = 250;    // 1000 f32 == exactly 250 float4 (4000 B)
constexpr int   kBlock = 256;    // 8 wave32 per block
constexpr int   kWaves = 8;
constexpr int   kGrid  = 2048;   // 16384 waves -> 8 rows/wave at B=131072
constexpr int   kBufF  = 1024;   // floats per LDS buffer (4 KB, 16B aligned)
}

typedef float f4raw __attribute__((ext_vector_type(4)));  // native vector for NT stores

// Issue the async copy of one 4000-byte row into this wave's LDS buffer.
// 32 lanes x 16B = 512B per instruction; 7 full instructions + 1 EXEC-masked
// tail (chunk c = 224+lane valid only while c < 250 i.e. lane < 26).
// Exactly 8 ASYNCcnt increments per row for every wave (EXEC never 0).
// TH_LOAD_NT: streaming data, no reuse -> don't let it churn L2.
__device__ __forceinline__ void issue_async_row(const float* __restrict__ rowPtr,
                                                uint32_t ldsByteBase, int lane) {
    uint64_t g = (uint64_t)(uintptr_t)rowPtr + (uint64_t)(lane * 16);
    uint32_t l = ldsByteBase + (uint32_t)(lane * 16);
    asm volatile("global_load_async_to_lds_b128 %0, %1, off th:TH_LOAD_NT"             :: "v"(l), "v"(g) : "memory");
    asm volatile("global_load_async_to_lds_b128 %0, %1, off offset:512 th:TH_LOAD_NT"  :: "v"(l), "v"(g) : "memory");
    asm volatile("global_load_async_to_lds_b128 %0, %1, off offset:1024 th:TH_LOAD_NT" :: "v"(l), "v"(g) : "memory");
    asm volatile("global_load_async_to_lds_b128 %0, %1, off offset:1536 th:TH_LOAD_NT" :: "v"(l), "v"(g) : "memory");
    asm volatile("global_load_async_to_lds_b128 %0, %1, off offset:2048 th:TH_LOAD_NT" :: "v"(l), "v"(g) : "memory");
    asm volatile("global_load_async_to_lds_b128 %0, %1, off offset:2560 th:TH_LOAD_NT" :: "v"(l), "v"(g) : "memory");
    asm volatile("global_load_async_to_lds_b128 %0, %1, off offset:3072 th:TH_LOAD_NT" :: "v"(l), "v"(g) : "memory");
    if (lane < (kRowF4 - 224)) { // lanes 0..25 cover bytes 3584..3999
        asm volatile("global_load_async_to_lds_b128 %0, %1, off offset:3584 th:TH_LOAD_NT" :: "v"(l), "v"(g) : "memory");
    }
}

__global__ __launch_bounds__(kBlock)
void dkc_kernel(const float* __restrict__ in, float* __restrict__ out, int nrows) {
    // 8 waves * 2 buffers * 4 KB = 64 KB static LDS per workgroup (WGP has 320 KB).
    __shared__ __align__(16) float lds[kWaves][2][kBufF];

    const int lane = (int)(threadIdx.x & 31u);
    const int wave = (int)(threadIdx.x >> 5);
    const int wid  = (int)blockIdx.x * kWaves + wave;  // global wave id = row owner
    const int nw   = (int)gridDim.x * kWaves;          // row stride

    // Workgroup-relative LDS byte addresses of this wave's two buffers
    // (low 32 bits of the flat LDS-aperture address == LDS byte offset).
    uint32_t ldsByte[2];
    ldsByte[0] = (uint32_t)(uintptr_t)&lds[wave][0][0];
    ldsByte[1] = (uint32_t)(uintptr_t)&lds[wave][1][0];

    int r = wid;
    if (r >= nrows) return;   // no barriers anywhere, so early-out is safe

    issue_async_row(in + (size_t)r * kC, ldsByte[0], lane);
    int buf = 0;

    while (r < nrows) {
        const int  rn   = r + nw;
        const bool more = rn < nrows;
        if (more) {
            // Prefetch next row into the other buffer (8 more ASYNCcnt).
            issue_async_row(in + (size_t)rn * kC, ldsByte[buf ^ 1], lane);
            // Async loads complete in order: <=8 outstanding => current row done.
            asm volatile("s_wait_asynccnt 0x8" ::: "memory");
        } else {
            asm volatile("s_wait_asynccnt 0x0" ::: "memory");
        }

        // ---- Pull this wave's row out of LDS (ds_load_b128, chunk c = i*32+lane) ----
        const float4* __restrict__ lrow = (const float4*)&lds[wave][buf][0];
        float4 v[8];
        #pragma unroll
        for (int i = 0; i < 8; ++i) {
            const int c = i * 32 + lane;
            v[i] = lrow[c < kRowF4 ? c : 0];
        }

        // ---- Row reduction: max (+ first argmax index) and min ----
        float vmax = -__builtin_huge_valf();
        float vmin =  __builtin_huge_valf();
        int   vidx = 0x7fffffff;
        #pragma unroll
        for (int i = 0; i < 8; ++i) {
            const int c = i * 32 + lane;
            if (c < kRowF4) {
                const int b = 4 * c;
                const float x0 = v[i].x, x1 = v[i].y, x2 = v[i].z, x3 = v[i].w;
                if (x0 > vmax) { vmax = x0; vidx = b + 0; }
                if (x1 > vmax) { vmax = x1; vidx = b + 1; }
                if (x2 > vmax) { vmax = x2; vidx = b + 2; }
                if (x3 > vmax) { vmax = x3; vidx = b + 3; }
                vmin = fminf(vmin, fminf(fminf(x0, x1), fminf(x2, x3)));
            }
        }
        // wave32 butterfly; tie-break to smallest index (matches jnp.argmax).
        #pragma unroll
        for (int m = 16; m > 0; m >>= 1) {
            const float omax = __shfl_xor(vmax, m, 32);
            const int   oidx = __shfl_xor(vidx, m, 32);
            const float omin = __shfl_xor(vmin, m, 32);
            if (omax > vmax || (omax == vmax && oidx < vidx)) { vmax = omax; vidx = oidx; }
            vmin = fminf(vmin, omin);
        }

        // ---- Per-row scalars ----
        const float a    = (float)kC * kT - 1.0f;       // 899
        const float den  = (float)kC * vmax - 1.0f;     // 1000*max - 1
        const float invd = 1.0f / den;                  // one precise divide per row
        const float bofs = vmax - kT;
        // min(new_k) attained at row min or row max (affine in x, any slope sign)
        const float nkmin = fminf(fmaf(a, vmin, bofs) * invd,
                                  fmaf(a, vmax, bofs) * invd);
        const bool  fb  = nkmin < 0.0f;
        const float uni = (1.0f - kT) / (float)(kC - 1);

        // ---- Emit: 250 aligned non-temporal b128 stores per row ----
        f4raw* __restrict__ orow = (f4raw*)(out + (size_t)r * kC);
        #pragma unroll
        for (int i = 0; i < 8; ++i) {
            const int c = i * 32 + lane;
            if (c < kRowF4) {
                const int b = 4 * c;
                f4raw o;
                if (fb) {
                    o.x = (b + 0 == vidx) ? kT : uni;
                    o.y = (b + 1 == vidx) ? kT : uni;
                    o.z = (b + 2 == vidx) ? kT : uni;
                    o.w = (b + 3 == vidx) ? kT : uni;
                } else {
                    o.x = fmaf(a, v[i].x, bofs) * invd;
                    o.y = fmaf(a, v[i].y, bofs) * invd;
                    o.z = fmaf(a, v[i].z, bofs) * invd;
                    o.w = fmaf(a, v[i].w, bofs) * invd;
                }
                __builtin_nontemporal_store(o, &orow[c]);
            }
        }

        r = rn;
        buf ^= 1;
    }
}

extern "C" void kernel_launch(void* const* d_in, const int* in_sizes, int n_in,
                              void* d_out, int out_size, void* d_ws, size_t ws_size,
                              hipStream_t stream) {
    (void)n_in; (void)out_size; (void)d_ws; (void)ws_size;
    const float* logits = (const float*)d_in[0];
    float*       out    = (float*)d_out;
    const int nrows = in_sizes[0] / kC;   // 131072

    int grid = kGrid;
    const int maxBlocks = (nrows + kWaves - 1) / kWaves; // never launch idle-only blocks
    if (grid > maxBlocks) grid = maxBlocks;
    if (grid < 1) grid = 1;

    dkc_kernel<<<grid, kBlock, 0, stream>>>(logits, out, nrows);
}